// DPRNNUnit_33526514712780
// MI455X (gfx1250) — compile-verified
//
#include <hip/hip_runtime.h>
#include <math.h>

// Problem constants (from reference)
#define B_   4
#define C_   64
#define F_   128
#define T_   128
#define K_   8
#define H_   32
#define S_   121   // T - K + 1
#define N_   512   // B * F
#define CK_  512   // C * K
#define G4_  128   // 4 * H
#define H2_  64    // 2 * H

typedef float v2f __attribute__((ext_vector_type(2)));
typedef float v8f __attribute__((ext_vector_type(8)));
typedef unsigned int u32x4 __attribute__((ext_vector_type(4)));
typedef int i32x4 __attribute__((ext_vector_type(4)));
typedef int i32x8 __attribute__((ext_vector_type(8)));

#ifndef __has_builtin
#define __has_builtin(x) 0
#endif
#if __has_builtin(__builtin_amdgcn_tensor_load_to_lds) && \
    __has_builtin(__builtin_amdgcn_s_wait_tensorcnt)
#define USE_TDM 1
#else
#define USE_TDM 0
#endif

__device__ __forceinline__ float sigm(float x) { return 1.0f / (1.0f + __expf(-x)); }

// fp32 WMMA wrapper: D = A(16x4) * B(4x16) + C   (wave32, 2 VGPRs A/B, 8 VGPRs C/D)
__device__ __forceinline__ v8f wmma_f32(v2f a, v2f b, v8f c) {
    return __builtin_amdgcn_wmma_f32_16x16x4_f32(false, a, false, b, (short)0, c,
                                                 false, false);
}

#if USE_TDM
// TDM: DMA a contiguous 16 KB (4096 x fp32, 1-D tensor == tile) global -> LDS.
__device__ __forceinline__ void tdm_load_16kb(const float* gsrc, float* lds_dst) {
    unsigned long long ga = (unsigned long long)(size_t)gsrc;
    unsigned lds_off = (unsigned)(size_t)lds_dst;  // flat LDS addr truncates to offset
    u32x4 g0;
    g0.x = 1u;                                          // count=1 (valid user D#)
    g0.y = lds_off;                                     // lds_addr (bytes)
    g0.z = (unsigned)(ga & 0xFFFFFFFFu);                // global_addr[31:0]
    g0.w = (unsigned)((ga >> 32) & 0x1FFFFFFu)          // global_addr[56:32]
         | (2u << 30);                                  // type = 2 ("image")
    i32x8 g1;
    g1[0] = 0x00020000;                 // workgroup_mask=0, data_size=2 (4 bytes)
    g1[1] = (int)(4096u << 16);         // tensor_dim0[15:0]=4096 at bits[63:48]
    g1[2] = 0x00010000;                 // tensor_dim0[31:16]=0 | tensor_dim1=1
    g1[3] = (int)(4096u << 16);         // tile_dim0 = 4096
    g1[4] = 0;                          // tile_dim1/2 = 0 (unused)
    g1[5] = 4096;                       // tensor_dim0_stride = 4096
    g1[6] = 0;
    g1[7] = 0;
    i32x4 z4 = {0, 0, 0, 0};
#if __clang_major__ >= 23
    i32x8 z8 = {0, 0, 0, 0, 0, 0, 0, 0};
    __builtin_amdgcn_tensor_load_to_lds(g0, g1, z4, z4, z8, 0);
#else
    __builtin_amdgcn_tensor_load_to_lds(g0, g1, z4, z4, 0);
#endif
}
#endif

// ---------------------------------------------------------------------------
// Kernel 1: InstanceNorm2d over (T,F) per (b,c); write xf in [BF, C, T] layout
// ---------------------------------------------------------------------------
__global__ void __launch_bounds__(256) k_instnorm(const float* __restrict__ x,
                                                  const float* __restrict__ gamma,
                                                  const float* __restrict__ beta,
                                                  float* __restrict__ xf) {
    int bc = blockIdx.x;
    int b = bc / C_, c = bc % C_;
    const float* xp = x + (size_t)bc * (F_ * T_);
    float s = 0.f, s2 = 0.f;
    for (int i = threadIdx.x; i < F_ * T_; i += 256) {
        float v = xp[i];
        s += v; s2 += v * v;
    }
    __shared__ float rs[256], rq[256];
    rs[threadIdx.x] = s; rq[threadIdx.x] = s2;
    __syncthreads();
    for (int off = 128; off > 0; off >>= 1) {
        if (threadIdx.x < (unsigned)off) {
            rs[threadIdx.x] += rs[threadIdx.x + off];
            rq[threadIdx.x] += rq[threadIdx.x + off];
        }
        __syncthreads();
    }
    const float inv = 1.0f / (float)(F_ * T_);
    float mu  = rs[0] * inv;
    float var = rq[0] * inv - mu * mu;
    float sc  = gamma[c] * rsqrtf(var + 1e-5f);
    float sh  = beta[c] - mu * sc;
    for (int i = threadIdx.x; i < F_ * T_; i += 256) {
        int f = i / T_, t = i % T_;
        xf[((size_t)(b * F_ + f) * C_ + c) * T_ + t] = xp[i] * sc + sh;
    }
}

// ---------------------------------------------------------------------------
// Kernel 2: layer-0 input projection.  gx[d,s,n,g] = seq[s,n,:] . wih0[d,g,:] + bias
// seq[s,n, c*8+k] = xf[n, c, s+k].  One wave -> 16(n) x 64(g): A fragment reused
// across 4 WMMAs (1.25 b64 loads per WMMA instead of 2).
// ---------------------------------------------------------------------------
__global__ void __launch_bounds__(256) k_gx0(const float* __restrict__ xf,
                                             const float* __restrict__ wih,
                                             const float* __restrict__ bih,
                                             const float* __restrict__ bhh,
                                             float* __restrict__ gx) {
    int wid  = blockIdx.x * 8 + (threadIdx.x >> 5);
    int lane = threadIdx.x & 31;
    int gq = wid & 1;  wid >>= 1;       // 0..1 : which 64-wide gate slab
    int nt = wid & 31; wid >>= 5;       // 0..31: 16-row tile
    int s  = wid % S_;
    int d  = wid / S_;
    int row = lane & 15, half = lane >> 4;
    int n0 = nt * 16, g0 = gq * 64;

    const float* Ap  = xf + (size_t)(n0 + row) * (C_ * T_) + s;    // + c*T + k
    const float* Bp0 = wih + (size_t)d * G4_ * CK_ + (size_t)(g0 + row) * CK_;

    v8f z = {0.f, 0.f, 0.f, 0.f, 0.f, 0.f, 0.f, 0.f};
    v8f acc[4] = {z, z, z, z};
#pragma unroll 2
    for (int i0 = 0; i0 < CK_; i0 += 4) {
        int ia = i0 + half * 2;              // K layout: v0 -> K0/K2, v1 -> K1/K3
        float a0 = Ap[(ia >> 3) * T_ + (ia & 7)];
        float a1 = Ap[((ia + 1) >> 3) * T_ + ((ia + 1) & 7)];
        v2f Av = {a0, a1};
#pragma unroll
        for (int u = 0; u < 4; ++u) {
            const float* Bp = Bp0 + (size_t)u * 16 * CK_;
            v2f Bv = {Bp[ia], Bp[ia + 1]};
            acc[u] = wmma_f32(Av, Bv, acc[u]);
        }
    }
    float* out = gx + ((size_t)d * S_ + s) * (size_t)N_ * G4_;
#pragma unroll
    for (int u = 0; u < 4; ++u) {
        int gcol = g0 + u * 16 + row;
        float bias = bih[d * G4_ + gcol] + bhh[d * G4_ + gcol];
#pragma unroll
        for (int j = 0; j < 8; ++j) {
            int rn = n0 + j + half * 8;      // D layout: vgpr j -> M=j / j+8
            out[(size_t)rn * G4_ + gcol] = acc[u][j] + bias;
        }
    }
}

// ---------------------------------------------------------------------------
// Kernel 3: input projection for layers >= 1 (A contiguous, K = 2H = 64),
// same 16x64 register blocking.
// ---------------------------------------------------------------------------
__global__ void __launch_bounds__(256) k_gx1(const float* __restrict__ hin,
                                             const float* __restrict__ wih,
                                             const float* __restrict__ bih,
                                             const float* __restrict__ bhh,
                                             float* __restrict__ gx) {
    int wid  = blockIdx.x * 8 + (threadIdx.x >> 5);
    int lane = threadIdx.x & 31;
    int gq = wid & 1;  wid >>= 1;
    int nt = wid & 31; wid >>= 5;
    int s  = wid % S_;
    int d  = wid / S_;
    int row = lane & 15, half = lane >> 4;
    int n0 = nt * 16, g0 = gq * 64;

    const float* Ap  = hin + ((size_t)s * N_ + n0 + row) * H2_;
    const float* Bp0 = wih + (size_t)d * G4_ * H2_ + (size_t)(g0 + row) * H2_;

    v8f z = {0.f, 0.f, 0.f, 0.f, 0.f, 0.f, 0.f, 0.f};
    v8f acc[4] = {z, z, z, z};
#pragma unroll
    for (int i0 = 0; i0 < H2_; i0 += 4) {
        int ia = i0 + half * 2;
        v2f Av = {Ap[ia], Ap[ia + 1]};
#pragma unroll
        for (int u = 0; u < 4; ++u) {
            const float* Bp = Bp0 + (size_t)u * 16 * H2_;
            v2f Bv = {Bp[ia], Bp[ia + 1]};
            acc[u] = wmma_f32(Av, Bv, acc[u]);
        }
    }
    float* out = gx + ((size_t)d * S_ + s) * (size_t)N_ * G4_;
#pragma unroll
    for (int u = 0; u < 4; ++u) {
        int gcol = g0 + u * 16 + row;
        float bias = bih[d * G4_ + gcol] + bhh[d * G4_ + gcol];
#pragma unroll
        for (int j = 0; j < 8; ++j) {
            int rn = n0 + j + half * 8;
            out[(size_t)rn * G4_ + gcol] = acc[u][j] + bias;
        }
    }
}

// ---------------------------------------------------------------------------
// Kernel 4: LSTM recurrence. Row n of g = gx[s] + h.Whh^T needs only row n of h,
// so each (direction, 16-row tile) is fully independent: 64 single-wave blocks.
// Whh staged to LDS via the Tensor Data Mover; next-step gx prefetched.
// ---------------------------------------------------------------------------
__global__ void __launch_bounds__(32) k_lstm(const float* __restrict__ gx,
                                             const float* __restrict__ whh2, // [2][128][32]
                                             float* __restrict__ hout) {    // [S][N][64]
    int d  = blockIdx.x >> 5;        // direction
    int mt = blockIdx.x & 31;        // row tile
    int lane = threadIdx.x;
    int row = lane & 15, half = lane >> 4;
    int n0 = mt * 16;

    __shared__ float whs[G4_ * H_];  // 16 KB
    __shared__ float hs[16 * H_];    // 2 KB
    __shared__ float cs[16 * H_];    // 2 KB

    const float* wp = whh2 + (size_t)d * G4_ * H_;
#if USE_TDM
    tdm_load_16kb(wp, whs);                      // one TDM DMA per wave
#else
    for (int i = lane; i < G4_ * H_; i += 32) whs[i] = wp[i];
#endif
    for (int i = lane; i < 16 * H_; i += 32) { hs[i] = 0.f; cs[i] = 0.f; }
#if USE_TDM
    __builtin_amdgcn_s_wait_tensorcnt(0);        // Whh resident before first use
#endif
    __syncthreads();

    const float* gxd = gx + (size_t)d * S_ * (size_t)N_ * G4_;
    for (int ss = 0; ss < S_; ++ss) {
        int s = d ? (S_ - 1 - ss) : ss;
        const float* gp = gxd + ((size_t)s * N_ + n0) * G4_;

        // prefetch next step's 8 KB gx slice (32 lanes x 256 B covers it)
        if (ss + 1 < S_) {
            int sn = d ? (S_ - 2 - ss) : (ss + 1);
            const float* pf = gxd + ((size_t)sn * N_ + n0) * G4_ + (size_t)lane * 64;
            __builtin_prefetch((const void*)pf, 0, 1);
        }

        v8f acc[8];
#pragma unroll
        for (int gt = 0; gt < 8; ++gt) {         // init C from gx (D/C layout)
            v8f a;
#pragma unroll
            for (int j = 0; j < 8; ++j)
                a[j] = gp[(size_t)(j + half * 8) * G4_ + gt * 16 + row];
            acc[gt] = a;
        }
#pragma unroll
        for (int k0 = 0; k0 < H_; k0 += 4) {     // g += h . Whh^T
            v2f Av = { hs[row * H_ + k0 + half * 2],
                       hs[row * H_ + k0 + half * 2 + 1] };
#pragma unroll
            for (int gt = 0; gt < 8; ++gt) {
                v2f Bv = { whs[(gt * 16 + row) * H_ + k0 + half * 2],
                           whs[(gt * 16 + row) * H_ + k0 + half * 2 + 1] };
                acc[gt] = wmma_f32(Av, Bv, acc[gt]);
            }
        }
        // gates: i = tiles 0..1, f = 2..3, g = 4..5, o = 6..7 (same vgpr/lane slot)
        float* hop = hout + ((size_t)s * N_ + n0) * H2_ + d * H_;
#pragma unroll
        for (int p = 0; p < 2; ++p) {
            int jh = p * 16 + row;               // hidden index
#pragma unroll
            for (int j = 0; j < 8; ++j) {
                int nl = j + half * 8;           // local row
                float iv = sigm(acc[p][j]);
                float fv = sigm(acc[2 + p][j]);
                float gv = tanhf(acc[4 + p][j]);
                float ov = sigm(acc[6 + p][j]);
                float cv = fv * cs[nl * H_ + jh] + iv * gv;
                float hv = ov * tanhf(cv);
                cs[nl * H_ + jh] = cv;
                hs[nl * H_ + jh] = hv;
                hop[(size_t)nl * H2_ + jh] = hv;
            }
        }
        __syncthreads();                         // single-wave: orders LDS ops
    }
}

// ---------------------------------------------------------------------------
// Kernel 5: ConvTranspose1d (stride 1, VALID) as gathered GEMM + bias + residual
// y[n,cout,t] = sum_{cin,k} h[t-k, n, cin] * wct[cin, cout, k]  (0 <= t-k < S)
// One wave -> 16(t) x 64(cout): A reused across all 4 cout tiles.
// ---------------------------------------------------------------------------
__global__ void __launch_bounds__(256) k_deconv(const float* __restrict__ hfin, // [S][N][64]
                                                const float* __restrict__ wct,  // [64][C][K]
                                                const float* __restrict__ bct,
                                                const float* __restrict__ x,    // residual
                                                float* __restrict__ out) {
    int wid  = blockIdx.x * 8 + (threadIdx.x >> 5);
    int lane = threadIdx.x & 31;
    int tt = wid & 7;                  // t tile (T/16 = 8)
    int n  = wid >> 3;                 // 0..511
    int row = lane & 15, half = lane >> 4;
    int t0 = tt * 16;

    v8f z = {0.f, 0.f, 0.f, 0.f, 0.f, 0.f, 0.f, 0.f};
    v8f acc[4] = {z, z, z, z};
#pragma unroll 2
    for (int i0 = 0; i0 < H2_ * K_; i0 += 4) {
        int ia = i0 + half * 2;
        float a0 = 0.f, a1 = 0.f;
        {
            int cin = ia >> 3, k = ia & 7, ts = t0 + row - k;
            if (ts >= 0 && ts < S_) a0 = hfin[((size_t)ts * N_ + n) * H2_ + cin];
        }
        {
            int ib = ia + 1;
            int cin = ib >> 3, k = ib & 7, ts = t0 + row - k;
            if (ts >= 0 && ts < S_) a1 = hfin[((size_t)ts * N_ + n) * H2_ + cin];
        }
        v2f Av = {a0, a1};
#pragma unroll
        for (int u = 0; u < 4; ++u) {
            int col = u * 16 + row;
            float b0 = wct[((size_t)(ia >> 3) * C_ + col) * K_ + (ia & 7)];
            float b1 = wct[((size_t)((ia + 1) >> 3) * C_ + col) * K_ + ((ia + 1) & 7)];
            v2f Bv = {b0, b1};
            acc[u] = wmma_f32(Av, Bv, acc[u]);
        }
    }
    int b = n >> 7, f = n & 127;
#pragma unroll
    for (int u = 0; u < 4; ++u) {
        int cout = u * 16 + row;
        float bias = bct[cout];
#pragma unroll
        for (int j = 0; j < 8; ++j) {
            int t = t0 + j + half * 8;
            size_t oidx = (((size_t)b * C_ + cout) * F_ + f) * T_ + t;
            out[oidx] = acc[u][j] + bias + x[oidx];
        }
    }
}

// ---------------------------------------------------------------------------
// Workspace layout (floats):
//   xf : N*C*T     = 4,194,304   (16.0 MB)
//   gx : 2*S*N*4H  = 15,859,712  (63.4 MB)
//   h0 : S*N*2H    = 3,964,928   (15.9 MB)
//   h1 : S*N*2H    = 3,964,928   (15.9 MB)
// total ~112 MB
// ---------------------------------------------------------------------------
extern "C" void kernel_launch(void* const* d_in, const int* in_sizes, int n_in,
                              void* d_out, int out_size, void* d_ws, size_t ws_size,
                              hipStream_t stream) {
    const float* x     = (const float*)d_in[0];
    const float* gamma = (const float*)d_in[1];
    const float* beta  = (const float*)d_in[2];
    const float* w_ih0 = (const float*)d_in[3];
    const float* w_hh0 = (const float*)d_in[4];
    const float* b_ih0 = (const float*)d_in[5];
    const float* b_hh0 = (const float*)d_in[6];
    const float* w_ih  = (const float*)d_in[7];
    const float* w_hh  = (const float*)d_in[8];
    const float* b_ih  = (const float*)d_in[9];
    const float* b_hh  = (const float*)d_in[10];
    const float* wct   = (const float*)d_in[11];
    const float* bct   = (const float*)d_in[12];
    float* out = (float*)d_out;

    float* xf = (float*)d_ws;
    float* gx = xf + (size_t)N_ * C_ * T_;
    float* h0 = gx + (size_t)2 * S_ * N_ * G4_;
    float* h1 = h0 + (size_t)S_ * N_ * H2_;

    k_instnorm<<<B_ * C_, 256, 0, stream>>>(x, gamma, beta, xf);

    const int GX_BLOCKS = (2 * S_ * 32 * 2) / 8;   // 1936 blocks, 8 waves each
    k_gx0<<<GX_BLOCKS, 256, 0, stream>>>(xf, w_ih0, b_ih0, b_hh0, gx);
    k_lstm<<<64, 32, 0, stream>>>(gx, w_hh0, h0);

    float* hin = h0;
    float* ho  = h1;
    for (int l = 1; l < 4; ++l) {
        const float* wih_l = w_ih + (size_t)(l - 1) * 2 * G4_ * H2_;
        const float* whh_l = w_hh + (size_t)(l - 1) * 2 * G4_ * H_;
        const float* bih_l = b_ih + (size_t)(l - 1) * 2 * G4_;
        const float* bhh_l = b_hh + (size_t)(l - 1) * 2 * G4_;
        k_gx1<<<GX_BLOCKS, 256, 0, stream>>>(hin, wih_l, bih_l, bhh_l, gx);
        k_lstm<<<64, 32, 0, stream>>>(gx, whh_l, ho);
        float* t = hin; hin = ho; ho = t;
    }

    k_deconv<<<(N_ * 8) / 8, 256, 0, stream>>>(hin, wct, bct, x, out);
}